// VariationalGCNEncoder_3032246911086
// MI455X (gfx1250) — compile-verified
//
#include <hip/hip_runtime.h>

#define IN_CH 128
#define HID    64
#define OUT_CH 32

typedef __attribute__((ext_vector_type(2))) float v2f;
typedef __attribute__((ext_vector_type(8))) float v8f;

// ---------------------------------------------------------------------------
// fp32 scatter-add: force the native L2 atomic (global_atomic_add_f32),
// not a CAS loop. Working set fits in the 192MB L2, so these are L2-latency.
// ---------------------------------------------------------------------------
__device__ __forceinline__ void atomAddF(float* p, float v) {
    unsafeAtomicAdd(p, v);
}

// ---------------------------------------------------------------------------
// Degree accumulation: deg[dst] += 1 over the 800K explicit edges.
// ---------------------------------------------------------------------------
__global__ void deg_kernel(const int* __restrict__ dst, float* __restrict__ deg, int E) {
    int e = blockIdx.x * blockDim.x + threadIdx.x;
    if (e < E) atomAddF(deg + dst[e], 1.0f);
}

// dinv[i] = rsqrt(deg[i] + 1)   (+1 = self loop)
__global__ void dinv_kernel(const float* __restrict__ deg, float* __restrict__ dinv, int N) {
    int i = blockIdx.x * blockDim.x + threadIdx.x;
    if (i < N) dinv[i] = __frsqrt_rn(deg[i] + 1.0f);
}

// ---------------------------------------------------------------------------
// GEMM1: hpre[N x 64] = X[N x 128] @ W1[128 x 64], fp32 WMMA 16x16x4.
// One wave per 16-row strip; 4 accumulators cover all 64 output columns,
// A fragment reused across the 4 WMMAs per K-step.
// ---------------------------------------------------------------------------
__global__ void gemm1_kernel(const float* __restrict__ X, const float* __restrict__ W1,
                             float* __restrict__ hpre, int tilesM) {
    const int wid   = threadIdx.x >> 5;
    const int lane  = threadIdx.x & 31;
    const int tileM = blockIdx.x * (blockDim.x >> 5) + wid;
    if (tileM >= tilesM) return;              // whole-wave guard: EXEC stays all-1s

    const int m    = lane & 15;               // row (A) / col (B,C,D) within tile
    const int half = lane >> 4;               // K pair select for A/B fragments

    const float* arow = X + (size_t)(tileM * 16 + m) * IN_CH;

    v8f acc0 = {}, acc1 = {}, acc2 = {}, acc3 = {};

    #pragma unroll 4
    for (int k = 0; k < IN_CH; k += 4) {
        const float2 av = *(const float2*)(arow + k + half * 2);
        v2f a; a.x = av.x; a.y = av.y;

        const float* b0 = W1 + (size_t)(k + half * 2    ) * HID;
        const float* b1 = W1 + (size_t)(k + half * 2 + 1) * HID;
        v2f bA; bA.x = b0[m     ]; bA.y = b1[m     ];
        v2f bB; bB.x = b0[m + 16]; bB.y = b1[m + 16];
        v2f bC; bC.x = b0[m + 32]; bC.y = b1[m + 32];
        v2f bD; bD.x = b0[m + 48]; bD.y = b1[m + 48];

        acc0 = __builtin_amdgcn_wmma_f32_16x16x4_f32(false, a, false, bA, (short)0, acc0, false, false);
        acc1 = __builtin_amdgcn_wmma_f32_16x16x4_f32(false, a, false, bB, (short)0, acc1, false, false);
        acc2 = __builtin_amdgcn_wmma_f32_16x16x4_f32(false, a, false, bC, (short)0, acc2, false, false);
        acc3 = __builtin_amdgcn_wmma_f32_16x16x4_f32(false, a, false, bD, (short)0, acc3, false, false);
    }

    // C/D layout: VGPR v -> row = (lane>>4)*8 + v, col = lane&15 (+ tile col offset)
    const int crow = tileM * 16 + half * 8;
    #pragma unroll
    for (int v = 0; v < 8; ++v) {
        float* r = hpre + (size_t)(crow + v) * HID + m;
        r[0]  = acc0[v];
        r[16] = acc1[v];
        r[32] = acc2[v];
        r[48] = acc3[v];
    }
}

// ---------------------------------------------------------------------------
// Edge aggregation: one wave per edge, each lane owns 2 channels (float2) of
// the 64-channel feature row. out[dst] += norm * feat[src].
// ---------------------------------------------------------------------------
__global__ void agg_kernel(const float* __restrict__ feat, const int* __restrict__ src,
                           const int* __restrict__ dst, const float* __restrict__ dinv,
                           float* __restrict__ out, int E) {
    const int wid = threadIdx.x >> 5;
    const int e   = blockIdx.x * (blockDim.x >> 5) + wid;
    if (e >= E) return;
    const int lane = threadIdx.x & 31;

    const int s = src[e];
    const int d = dst[e];
    const float norm = dinv[s] * dinv[d];

    const float2 v = *(const float2*)(feat + (size_t)s * HID + 2 * lane);
    float* o = out + (size_t)d * HID + 2 * lane;
    atomAddF(o,     v.x * norm);
    atomAddF(o + 1, v.y * norm);
}

// h = relu(agg1 + dinv^2 * hpre + b1), written in place into agg1
__global__ void relu_kernel(float* __restrict__ agg1, const float* __restrict__ hpre,
                            const float* __restrict__ dinv, const float* __restrict__ b1,
                            int N) {
    int idx = blockIdx.x * blockDim.x + threadIdx.x;
    if (idx >= N * HID) return;
    const int i = idx >> 6;          // /64
    const int c = idx & 63;
    const float di = dinv[i];
    const float v  = agg1[idx] + di * di * hpre[idx] + b1[c];
    agg1[idx] = fmaxf(v, 0.0f);
}

// ---------------------------------------------------------------------------
// GEMM2 (fused mu/logstd): pre2[N x 64] = h[N x 64] @ [W_mu | W_logstd][64 x 64].
// Column fragments 0-1 come from W_mu, 2-3 from W_logstd (uniform per fragment).
// ---------------------------------------------------------------------------
__global__ void gemm2_kernel(const float* __restrict__ H, const float* __restrict__ Wmu,
                             const float* __restrict__ Wls, float* __restrict__ pre2,
                             int tilesM) {
    const int wid   = threadIdx.x >> 5;
    const int lane  = threadIdx.x & 31;
    const int tileM = blockIdx.x * (blockDim.x >> 5) + wid;
    if (tileM >= tilesM) return;

    const int m    = lane & 15;
    const int half = lane >> 4;

    const float* arow = H + (size_t)(tileM * 16 + m) * HID;

    v8f acc0 = {}, acc1 = {}, acc2 = {}, acc3 = {};

    #pragma unroll 4
    for (int k = 0; k < HID; k += 4) {
        const float2 av = *(const float2*)(arow + k + half * 2);
        v2f a; a.x = av.x; a.y = av.y;

        const float* m0 = Wmu + (size_t)(k + half * 2    ) * OUT_CH;
        const float* m1 = Wmu + (size_t)(k + half * 2 + 1) * OUT_CH;
        const float* l0 = Wls + (size_t)(k + half * 2    ) * OUT_CH;
        const float* l1 = Wls + (size_t)(k + half * 2 + 1) * OUT_CH;
        v2f bA; bA.x = m0[m     ]; bA.y = m1[m     ];   // cols  0..15  (mu)
        v2f bB; bB.x = m0[m + 16]; bB.y = m1[m + 16];   // cols 16..31  (mu)
        v2f bC; bC.x = l0[m     ]; bC.y = l1[m     ];   // cols 32..47  (logstd)
        v2f bD; bD.x = l0[m + 16]; bD.y = l1[m + 16];   // cols 48..63  (logstd)

        acc0 = __builtin_amdgcn_wmma_f32_16x16x4_f32(false, a, false, bA, (short)0, acc0, false, false);
        acc1 = __builtin_amdgcn_wmma_f32_16x16x4_f32(false, a, false, bB, (short)0, acc1, false, false);
        acc2 = __builtin_amdgcn_wmma_f32_16x16x4_f32(false, a, false, bC, (short)0, acc2, false, false);
        acc3 = __builtin_amdgcn_wmma_f32_16x16x4_f32(false, a, false, bD, (short)0, acc3, false, false);
    }

    const int crow = tileM * 16 + half * 8;
    #pragma unroll
    for (int v = 0; v < 8; ++v) {
        float* r = pre2 + (size_t)(crow + v) * HID + m;
        r[0]  = acc0[v];
        r[16] = acc1[v];
        r[32] = acc2[v];
        r[48] = acc3[v];
    }
}

// Finalize: add self-loop term + bias, split fused 64ch buffer into mu / logstd
__global__ void fin_kernel(const float* __restrict__ agg2, const float* __restrict__ pre2,
                           const float* __restrict__ dinv, const float* __restrict__ bmu,
                           const float* __restrict__ bls, float* __restrict__ out, int N) {
    int idx = blockIdx.x * blockDim.x + threadIdx.x;
    if (idx >= N * HID) return;
    const int i = idx >> 6;
    const int c = idx & 63;
    const float di = dinv[i];
    const float v  = agg2[idx] + di * di * pre2[idx];
    if (c < OUT_CH)
        out[(size_t)i * OUT_CH + c] = v + bmu[c];
    else
        out[(size_t)N * OUT_CH + (size_t)i * OUT_CH + (c - OUT_CH)] = v + bls[c - OUT_CH];
}

// ---------------------------------------------------------------------------
extern "C" void kernel_launch(void* const* d_in, const int* in_sizes, int n_in,
                              void* d_out, int out_size, void* d_ws, size_t ws_size,
                              hipStream_t stream) {
    const float* x   = (const float*)d_in[0];
    const int*   ei  = (const int*)  d_in[1];
    const float* W1  = (const float*)d_in[2];
    const float* b1  = (const float*)d_in[3];
    const float* Wmu = (const float*)d_in[4];
    const float* bmu = (const float*)d_in[5];
    const float* Wls = (const float*)d_in[6];
    const float* bls = (const float*)d_in[7];

    const int N = in_sizes[0] / IN_CH;   // 50000
    const int E = in_sizes[1] / 2;       // 800000
    const int* srcI = ei;                // edge_index[0]
    const int* dstI = ei + E;            // edge_index[1]

    // Workspace layout (floats): deg | dinv | hpre | agg1(->h) | pre2 | agg2
    float* ws   = (float*)d_ws;
    float* deg  = ws;
    float* dinv = deg  + N;
    float* hpre = dinv + N;
    float* agg1 = hpre + (size_t)N * HID;
    float* pre2 = agg1 + (size_t)N * HID;
    float* agg2 = pre2 + (size_t)N * HID;

    hipMemsetAsync(deg,  0, (size_t)N * sizeof(float), stream);
    hipMemsetAsync(agg1, 0, (size_t)N * HID * sizeof(float), stream);
    hipMemsetAsync(agg2, 0, (size_t)N * HID * sizeof(float), stream);

    deg_kernel <<<(E + 255) / 256, 256, 0, stream>>>(dstI, deg, E);
    dinv_kernel<<<(N + 255) / 256, 256, 0, stream>>>(deg, dinv, N);

    const int tilesM = N / 16;                         // 3125, exact
    gemm1_kernel<<<(tilesM + 7) / 8, 256, 0, stream>>>(x, W1, hpre, tilesM);
    agg_kernel  <<<(E + 7) / 8,      256, 0, stream>>>(hpre, srcI, dstI, dinv, agg1, E);
    relu_kernel <<<(N * HID + 255) / 256, 256, 0, stream>>>(agg1, hpre, dinv, b1, N);

    gemm2_kernel<<<(tilesM + 7) / 8, 256, 0, stream>>>(agg1, Wmu, Wls, pre2, tilesM);
    agg_kernel  <<<(E + 7) / 8,      256, 0, stream>>>(pre2, srcI, dstI, dinv, agg2, E);
    fin_kernel  <<<(N * HID + 255) / 256, 256, 0, stream>>>(agg2, pre2, dinv, bmu, bls,
                                                            (float*)d_out, N);
}